// QuantumRegression__gen140_65481071410797
// MI455X (gfx1250) — compile-verified
//
#include <hip/hip_runtime.h>

#define NW  12
#define NL  4
#define NS  4096          // 2^12 amplitudes
#define TPB 128           // 4 wave32 per workgroup, one sample per workgroup

typedef __attribute__((ext_vector_type(2))) float v2f;
typedef __attribute__((ext_vector_type(8))) float v8f;

// Gray decode: returns v such that v ^ (v>>1) == u  (linear, bijective)
__device__ __forceinline__ int gdec(int u) {
  u ^= u >> 1; u ^= u >> 2; u ^= u >> 4; u ^= u >> 8;
  return u;
}

// Load the A operands (16x4 f32 chunks of a 16x16 U, VGPR0: K=kb, VGPR1: K=kb+1)
// plus a pre-negated Im(U) copy (f32 WMMA has no A/B negate modifier).
__device__ __forceinline__ void load_A(const float* Urp, int lane,
                                       v2f Ar[4], v2f Aip[4], v2f Ain[4])
{
  const float* Uip = Urp + 256;
  const int Mr = lane & 15;
  const int kb = (lane < 16) ? 0 : 2;
#pragma unroll
  for (int kc = 0; kc < 4; ++kc) {
    int K = kc * 4 + kb;
    v2f a0 = { Urp[Mr * 16 + K], Urp[Mr * 16 + K + 1] };
    v2f a1 = { Uip[Mr * 16 + K], Uip[Mr * 16 + K + 1] };
    Ar[kc] = a0; Aip[kc] = a1; Ain[kc] = -a1;
  }
}

// 16 complex WMMAs computing Dr/Di = U x S for one 16x16-element call-group.
__device__ __forceinline__ void cplx_wmma(const v2f Ar[4], const v2f Aip[4],
                                          const v2f Ain[4],
                                          const v2f Br[4], const v2f Bi[4],
                                          v8f& Dr, v8f& Di)
{
#pragma unroll
  for (int kc = 0; kc < 4; ++kc) {
    // Dr += Ur*Sr - Ui*Si ; Di += Ur*Si + Ui*Sr
    Dr = __builtin_amdgcn_wmma_f32_16x16x4_f32(false, Ar[kc],  false, Br[kc], (short)0, Dr, false, false);
    Dr = __builtin_amdgcn_wmma_f32_16x16x4_f32(false, Ain[kc], false, Bi[kc], (short)0, Dr, false, false);
    Di = __builtin_amdgcn_wmma_f32_16x16x4_f32(false, Ar[kc],  false, Bi[kc], (short)0, Di, false, false);
    Di = __builtin_amdgcn_wmma_f32_16x16x4_f32(false, Aip[kc], false, Br[kc], (short)0, Di, false, false);
  }
}

// In-place pass: apply U to the k-subspace addressed s = G*GS + k*KS + n*NSTR.
// Each wave owns disjoint call-groups; all 256 elements of a group are read
// into B registers before D is written, so in-place is safe.
template<int GS, int KS, int NSTR>
__device__ __forceinline__ void wmma_pass(float* re, float* im,
                                          const float* Urp, int lane, int wave)
{
  v2f Ar[4], Aip[4], Ain[4];
  load_A(Urp, lane, Ar, Aip, Ain);
  const int kb = (lane < 16) ? 0 : 2;
  const int mb = (lane < 16) ? 0 : 8;
  const int n  = lane & 15;
#pragma unroll 1
  for (int G = wave; G < 16; G += (TPB / 32)) {
    int base = G * GS + n * NSTR;
    v2f Br[4], Bi[4];
#pragma unroll
    for (int kc = 0; kc < 4; ++kc) {
      int s = base + (kc * 4 + kb) * KS;
      if constexpr (KS == 1) {
        Br[kc] = *(const v2f*)&re[s];          // ds_load_b64 (s even)
        Bi[kc] = *(const v2f*)&im[s];
      } else {
        v2f br = { re[s], re[s + KS] };  Br[kc] = br;
        v2f bi = { im[s], im[s + KS] };  Bi[kc] = bi;
      }
    }
    v8f Dr = {0.f,0.f,0.f,0.f,0.f,0.f,0.f,0.f};
    v8f Di = {0.f,0.f,0.f,0.f,0.f,0.f,0.f,0.f};
    cplx_wmma(Ar, Aip, Ain, Br, Bi, Dr, Di);
    int sb = base + mb * KS;
#pragma unroll
    for (int j = 0; j < 8; ++j) { re[sb + j * KS] = Dr[j]; im[sb + j * KS] = Di[j]; }
  }
}

// Third pass (bits 8..11, s = k*256 + G*16 + n) fused with the CNOT chain:
// the chain CNOT(0,1)..CNOT(10,11) composes to psi'[v] = psi[v ^ (v>>1)], so
// each result element s is scattered to dst[gdec(s)] in the other buffer.
// U-scratch lives in the destination buffer -> barrier between A-load and use.
__device__ __forceinline__ void wmma_pass_scatter(const float* re, const float* im,
                                                  float* dre, float* dim,
                                                  const float* Urp, int lane, int wave)
{
  v2f Ar[4], Aip[4], Ain[4];
  load_A(Urp, lane, Ar, Aip, Ain);
  __syncthreads();   // all A registers read before any wave overwrites scratch
  const int kb = (lane < 16) ? 0 : 2;
  const int mb = (lane < 16) ? 0 : 8;
  const int n  = lane & 15;
#pragma unroll 1
  for (int G = wave; G < 16; G += (TPB / 32)) {
    int base = G * 16 + n;
    v2f Br[4], Bi[4];
#pragma unroll
    for (int kc = 0; kc < 4; ++kc) {
      int s = base + (kc * 4 + kb) * 256;
      v2f br = { re[s], re[s + 256] };  Br[kc] = br;
      v2f bi = { im[s], im[s + 256] };  Bi[kc] = bi;
    }
    v8f Dr = {0.f,0.f,0.f,0.f,0.f,0.f,0.f,0.f};
    v8f Di = {0.f,0.f,0.f,0.f,0.f,0.f,0.f,0.f};
    cplx_wmma(Ar, Aip, Ain, Br, Bi, Dr, Di);
#pragma unroll
    for (int j = 0; j < 8; ++j) {
      int v = gdec(base + (mb + j) * 256);
      dre[v] = Dr[j]; dim[v] = Di[j];
    }
  }
}

// One workgroup simulates one sample's 12-qubit state in LDS (double buffered).
// Per layer: the 12 fused RZ*RX gates become three 16x16 unitaries applied by
// v_wmma_f32_16x16x4_f32 over bit-groups [0..3], [4..7], [8..11]; the CNOT
// chain (Gray permutation) is folded into pass 3's scattered writeback.
__global__ __launch_bounds__(TPB) void qsim12_wmma(
    const float* __restrict__ inputs,   // (B, 12)
    const float* __restrict__ params,   // (4, 12)
    const float* __restrict__ head_w,   // (1, 12)
    const float* __restrict__ head_b,   // (1,)
    float* __restrict__ out)            // (B,)
{
  __shared__ float lds[2][2][NS];       // [buf][re/im][idx] -> exactly 64 KB

  const int b    = blockIdx.x;
  const int tid  = threadIdx.x;
  const int lane = tid & 31;
  const int wave = tid >> 5;

  // ---------------- initial RY product state (real) into buf 0 ----------------
  {
    float* iang = &lds[1][0][0];        // buf1 is free: 24-float angle table
    if (tid < NW) {
      float s, c; sincosf(0.5f * inputs[b * NW + tid], &s, &c);
      iang[tid] = c; iang[NW + tid] = s;
    }
    __syncthreads();
    float cv[NW], sv[NW];
#pragma unroll
    for (int i = 0; i < NW; ++i) { cv[i] = iang[i]; sv[i] = iang[NW + i]; }
#pragma unroll 1
    for (int s = tid; s < NS; s += TPB) {
      float pr = 1.0f;
#pragma unroll
      for (int i = 0; i < NW; ++i)
        pr *= ((s >> (11 - i)) & 1) ? sv[i] : cv[i];
      lds[0][0][s] = pr;
      lds[0][1][s] = 0.0f;
    }
  }
  __syncthreads();

  int cur = 0;
#pragma unroll 1
  for (int l = 0; l < NL; ++l) {
    float* cre = &lds[cur][0][0];
    float* cim = &lds[cur][1][0];
    float* ore = &lds[cur ^ 1][0][0];
    float* oim = &lds[cur ^ 1][1][0];
    float* scratch = ore;               // 3 x (Ur 256 + Ui 256) floats
    float* ang     = ore + 1536;        // 24-float per-layer angle table

    // layer angles once per workgroup (uniform across threads)
    if (tid < NW) {
      float s, c; sincosf(0.5f * params[l * NW + tid], &s, &c);
      ang[tid] = c; ang[NW + tid] = s;
    }
    __syncthreads();

    // ---- build the three fused 16x16 complex unitaries ----
    // pass bit-offset beta: k-bit p <-> state bit (beta+p) <-> wire (11-beta-p)
#pragma unroll
    for (int pass = 0; pass < 3; ++pass) {
      float* Urp = scratch + pass * 512;
      float* Uip = Urp + 256;
      for (int e = tid; e < 256; e += TPB) {
        int x = e >> 4, y = e & 15;
        float pr = 1.0f, pi = 0.0f;
#pragma unroll
        for (int p = 0; p < 4; ++p) {
          int w = 11 - pass * 4 - p;    // compile-time after unrolling
          float c = ang[w], s = ang[NW + w];
          int xb = (x >> p) & 1, yb = (y >> p) & 1;
          // G = RZ(th)*RX(th):
          //  g00=(c^2,-c*s) g01=(-s^2,-s*c) g10=(s^2,-s*c) g11=(c^2,c*s)
          float gr, gi;
          if (xb == 0 && yb == 0)      { gr =  c * c; gi = -c * s; }
          else if (xb == 0)            { gr = -s * s; gi = -s * c; }
          else if (yb == 0)            { gr =  s * s; gi = -s * c; }
          else                         { gr =  c * c; gi =  c * s; }
          float nr = pr * gr - pi * gi;
          float ni = pr * gi + pi * gr;
          pr = nr; pi = ni;
        }
        Urp[e] = pr; Uip[e] = pi;
      }
    }
    __syncthreads();

    // ---- two in-place WMMA passes (bits 0..3, 4..7), then pass 3 (bits
    //      8..11) fused with the Gray permutation into the other buffer ----
    wmma_pass<256, 1, 16>(cre, cim, scratch,       lane, wave); // s = G*256+n*16+k
    __syncthreads();
    wmma_pass<256, 16, 1>(cre, cim, scratch + 512, lane, wave); // s = G*256+k*16+n
    __syncthreads();
    wmma_pass_scatter(cre, cim, ore, oim, scratch + 1024, lane, wave);
    __syncthreads();
    cur ^= 1;
  }

  // ---------------- readout (state in buf 0 after 4 flips, plain order) -------
  const float* fre = &lds[0][0][0];
  const float* fim = &lds[0][1][0];
  float* wsum = &lds[1][0][0];

  float w[NW];
#pragma unroll
  for (int i = 0; i < NW; ++i) w[i] = head_w[i];

  float acc = 0.0f;
#pragma unroll 1
  for (int s = tid; s < NS; s += TPB) {
    float p = fre[s] * fre[s] + fim[s] * fim[s];
    float coef = 0.0f;
#pragma unroll
    for (int i = 0; i < NW; ++i)
      coef += ((s >> (11 - i)) & 1) ? -w[i] : w[i];
    acc += p * coef;
  }
#pragma unroll
  for (int off = 16; off > 0; off >>= 1)
    acc += __shfl_xor(acc, off, 32);
  if (lane == 0) wsum[wave] = acc;
  __syncthreads();
  if (tid == 0) {
    float tot = 0.0f;
#pragma unroll
    for (int v = 0; v < TPB / 32; ++v) tot += wsum[v];
    out[b] = tot + head_b[0];
  }
}

extern "C" void kernel_launch(void* const* d_in, const int* in_sizes, int n_in,
                              void* d_out, int out_size, void* d_ws, size_t ws_size,
                              hipStream_t stream) {
  const float* inputs = (const float*)d_in[0];
  const float* params = (const float*)d_in[1];
  const float* head_w = (const float*)d_in[2];
  const float* head_b = (const float*)d_in[3];
  float* out = (float*)d_out;
  int batch = out_size;                 // 16384 samples, one per workgroup
  qsim12_wmma<<<batch, TPB, 0, stream>>>(inputs, params, head_w, head_b, out);
}